// genie_mmse_estimator_84499186581687
// MI455X (gfx1250) — compile-verified
//
#include <hip/hip_runtime.h>
#include <math.h>

// ---------------------------------------------------------------------------
// MMSE channel estimator, MI455X (gfx1250, wave32, WMMA).
//
// B=128, N=64, M=256, complex64. Dominant cost: 1000-iter GD loop =
// 2 complex 64^3 GEMMs / iter / batch (~537 GFLOP fp32) via
// v_wmma_f32_16x16x4_f32.
//   - C's A-fragments preloaded ONCE into VGPRs (constant across iterations;
//     global float4 loads deliver {re,im,re,im} = both complex fragments).
//   - B-side (x / P) kept TRANSPOSED in LDS so each B-fragment is a
//     contiguous 8-byte pair -> single ds_load_b64, no repack movs.
//   - eigh path collapses to inv(|p|^2 C + no I) -> complex Gauss-Jordan.
// ---------------------------------------------------------------------------

#define NB    128
#define ND    64
#define MO    256
#define ITERS 1000
#define LR    0.001f
#define LDP   65   // padded LDS leading dim -> conflict-free row access

typedef __attribute__((ext_vector_type(2))) float v2f;
typedef __attribute__((ext_vector_type(8))) float v8f;

__device__ __forceinline__ v8f wmma4(v2f a, v2f b, v8f c) {
  return __builtin_amdgcn_wmma_f32_16x16x4_f32(false, a, false, b, (short)0, c,
                                               false, false);
}

// B-fragment from a TRANSPOSED plane: B[k][c] = Mx[c][k]; pair is contiguous.
__device__ __forceinline__ v2f ldsBT(const float (*Mx)[LDP], int k0, int c0, int lane) {
  int c = c0 + (lane & 15);
  int k = k0 + ((lane >> 4) << 1);
  v2f b; b.x = Mx[c][k]; b.y = Mx[c][k + 1]; return b;
}

// A-fragment (16x4) straight from interleaved-complex global memory.
__device__ __forceinline__ void gA(const float* base, int row, int k,
                                   v2f& ar, v2f& ai) {
  float4 f = *(const float4*)(base + ((size_t)row * ND + k) * 2);
  ar.x = f.x; ar.y = f.z; ai.x = f.y; ai.y = f.w;
}

__device__ __forceinline__ float hash01(unsigned s) {
  s ^= s >> 17; s *= 0xed5ad4bbu;
  s ^= s >> 11; s *= 0xac4c1b51u;
  s ^= s >> 15; s *= 0x31848babu;
  s ^= s >> 14;
  return (float)(s >> 8) * (1.0f / 16777216.0f);
}

// ===========================================================================
// Kernel 1: Ainv[b] = inv(|p|^2 C[b] + no I)  (Hermitian PD complex GJ in LDS)
// ===========================================================================
extern "C" __global__ __launch_bounds__(64)
void k_herm_inverse(const float* __restrict__ Cg, const float* __restrict__ no,
                    const float* __restrict__ pilot, float* __restrict__ Ainv) {
  __shared__ float Ar[ND][LDP];
  __shared__ float Ai[ND][LDP];
  const int b = blockIdx.x;
  const int i = threadIdx.x;
  const float p2 = pilot[0] * pilot[0] + pilot[1] * pilot[1];
  const float nv = no[0];
  const float2* Cb = (const float2*)Cg + (size_t)b * ND * ND;
  #pragma unroll 8
  for (int j = 0; j < ND; ++j) {
    float2 c = Cb[i * ND + j];
    Ar[i][j] = p2 * c.x + (i == j ? nv : 0.0f);
    Ai[i][j] = p2 * c.y;
  }
  __syncthreads();
  for (int k = 0; k < ND; ++k) {
    if (i == k) {
      float pr = Ar[k][k], pmi = Ai[k][k];
      float d = pr * pr + pmi * pmi;
      float qr = pr / d, qi = -pmi / d;
      #pragma unroll 8
      for (int j = 0; j < ND; ++j) {
        float xr = Ar[k][j], xi = Ai[k][j];
        Ar[k][j] = xr * qr - xi * qi;
        Ai[k][j] = xr * qi + xi * qr;
      }
      Ar[k][k] = qr; Ai[k][k] = qi;
    }
    __syncthreads();
    if (i != k) {
      float fr = Ar[i][k], fi = Ai[i][k];
      float qr = Ar[k][k], qi = Ai[k][k];
      #pragma unroll 8
      for (int j = 0; j < ND; ++j) {
        if (j == k) continue;
        float br = Ar[k][j], bi = Ai[k][j];
        Ar[i][j] -= fr * br - fi * bi;
        Ai[i][j] -= fr * bi + fi * br;
      }
      Ar[i][k] = -(fr * qr - fi * qi);
      Ai[i][k] = -(fr * qi + fi * qr);
    }
    __syncthreads();
  }
  float2* Ob = (float2*)Ainv + (size_t)b * ND * ND;
  #pragma unroll 8
  for (int j = 0; j < ND; ++j) Ob[i * ND + j] = make_float2(Ar[i][j], Ai[i][j]);
}

// ===========================================================================
// Kernel 2: W[b] = conj(p) * C[b] @ Ainv[b]
// A (C) from global float4; B (Ainv) staged TRANSPOSED in LDS.
// ===========================================================================
extern "C" __global__ __launch_bounds__(256)
void k_mmse_filter(const float* __restrict__ Cg, const float* __restrict__ Ainv,
                   const float* __restrict__ pilot, float* __restrict__ W) {
  __shared__ float BreT[ND][LDP], BimT[ND][LDP];
  const int b = blockIdx.x;
  const int tid = threadIdx.x, wave = tid >> 5, lane = tid & 31;
  const float2* Ab = (const float2*)Ainv + (size_t)b * ND * ND;
  for (int e = tid; e < ND * ND; e += 256) {
    float2 a = Ab[e];                 // a = Ainv[i][j], i=e>>6, j=e&63
    BreT[e & 63][e >> 6] = a.x;       // transposed: BT[j][i]
    BimT[e & 63][e >> 6] = a.y;
  }
  __syncthreads();
  const int ti = (wave * 2) >> 2, tj0 = (wave * 2) & 3;
  const v8f z8 = {0, 0, 0, 0, 0, 0, 0, 0};
  v8f dr[2] = {z8, z8}, di[2] = {z8, z8};
  const float* Cbf = Cg + (size_t)b * ND * ND * 2;
  const int arow = ti * 16 + (lane & 15);
  const int koff = (lane >> 4) << 1;
  #pragma unroll 4
  for (int k0 = 0; k0 < ND; k0 += 4) {
    v2f ar, ai;
    gA(Cbf, arow, k0 + koff, ar, ai);
    v2f nai; nai.x = -ai.x; nai.y = -ai.y;   // F32 WMMA: no A/B NEG modifier
    #pragma unroll
    for (int t = 0; t < 2; ++t) {
      v2f br = ldsBT(BreT, k0, (tj0 + t) * 16, lane);
      v2f bi = ldsBT(BimT, k0, (tj0 + t) * 16, lane);
      dr[t] = wmma4(ar, br, dr[t]);
      dr[t] = wmma4(nai, bi, dr[t]);
      di[t] = wmma4(ar, bi, di[t]);
      di[t] = wmma4(ai, br, di[t]);
    }
  }
  const float pr = pilot[0], pi = pilot[1];  // conj(p) = pr - i*pi
  float2* Wb = (float2*)W + (size_t)b * ND * ND;
  const int rb = ti * 16 + ((lane >> 4) << 3);
  #pragma unroll
  for (int t = 0; t < 2; ++t) {
    int gn = (tj0 + t) * 16 + (lane & 15);
    #pragma unroll
    for (int v = 0; v < 8; ++v) {
      float xr = dr[t][v], xi = di[t][v];
      Wb[(rb + v) * ND + gn] = make_float2(pr * xr + pi * xi, pr * xi - pi * xr);
    }
  }
}

// ===========================================================================
// Kernel 3: h_hat[b] = y[b] (256x64) @ W[b]^T
// A (y) from global float4; B = W^T read from row-major W plane (contiguous).
// ===========================================================================
extern "C" __global__ __launch_bounds__(256)
void k_apply_filter(const float* __restrict__ y, const float* __restrict__ W,
                    float* __restrict__ hhat) {
  __shared__ float Wre[ND][LDP], Wim[ND][LDP];
  const int b = blockIdx.x;
  const int m0 = blockIdx.y * 64;
  const int tid = threadIdx.x, wave = tid >> 5, lane = tid & 31;
  const float2* Wb = (const float2*)W + (size_t)b * ND * ND;
  for (int e = tid; e < ND * ND; e += 256) {
    float2 w = Wb[e]; Wre[e >> 6][e & 63] = w.x; Wim[e >> 6][e & 63] = w.y;
  }
  __syncthreads();
  const int ti = (wave * 2) >> 2, tj0 = (wave * 2) & 3;
  const v8f z8 = {0, 0, 0, 0, 0, 0, 0, 0};
  v8f dr[2] = {z8, z8}, di[2] = {z8, z8};
  const float* yb = y + (size_t)b * MO * ND * 2;
  const int arow = m0 + ti * 16 + (lane & 15);
  const int koff = (lane >> 4) << 1;
  #pragma unroll 4
  for (int k0 = 0; k0 < ND; k0 += 4) {
    v2f ar, ai;
    gA(yb, arow, k0 + koff, ar, ai);
    v2f nai; nai.x = -ai.x; nai.y = -ai.y;
    #pragma unroll
    for (int t = 0; t < 2; ++t) {
      v2f br = ldsBT(Wre, k0, (tj0 + t) * 16, lane);  // B[k][n] = W[n][k]
      v2f bi = ldsBT(Wim, k0, (tj0 + t) * 16, lane);
      dr[t] = wmma4(ar, br, dr[t]);
      dr[t] = wmma4(nai, bi, dr[t]);
      di[t] = wmma4(ar, bi, di[t]);
      di[t] = wmma4(ai, br, di[t]);
    }
  }
  float2* Ob = (float2*)hhat + (size_t)b * MO * ND;
  const int rb = ti * 16 + ((lane >> 4) << 3);
  #pragma unroll
  for (int t = 0; t < 2; ++t) {
    int gn = (tj0 + t) * 16 + (lane & 15);
    #pragma unroll
    for (int v = 0; v < 8; ++v) {
      int gm = m0 + rb + v;
      Ob[(size_t)gm * ND + gn] = make_float2(dr[t][v], di[t][v]);
    }
  }
}

// ===========================================================================
// Kernel 4: GD approximate inverse (the 537-GFLOP hot loop).
// One block / batch. C's A-fragment strip lives in VGPRs for all 1000
// iterations (zero A-side loads in the loop); x lives in accumulator VGPRs;
// x / P bounce through ONE transposed LDS plane pair (33 KB total).
// Inner k-step: 4 contiguous ds_load_b64 + 8 v_wmma + 2 v_xor.
// Update: x -= (lr/K) * C @ ((Cx - I)/|Cx - I|), K = B*N*N (C Hermitian).
// ===========================================================================
extern "C" __global__ __launch_bounds__(256)
void k_gd_inverse(const float* __restrict__ Cg, float* __restrict__ Xout,
                  float scale) {
  __shared__ float XTre[ND][LDP], XTim[ND][LDP];  // transposed: XT[n][k]
  const int b = blockIdx.x;
  const int tid = threadIdx.x, wave = tid >> 5, lane = tid & 31;
  const int ti = (wave * 2) >> 2, tj0 = (wave * 2) & 3;
  const int rb = ti * 16 + ((lane >> 4) << 3);
  const int colL = lane & 15;
  const v8f z8 = {0, 0, 0, 0, 0, 0, 0, 0};

  // --- preload C A-fragment strip (rows ti*16..ti*16+15, all K) into VGPRs
  const float* Cbf = Cg + (size_t)b * ND * ND * 2;
  const int arow = ti * 16 + (lane & 15);
  const int koff = (lane >> 4) << 1;
  v2f car[16], cai[16];
  #pragma unroll
  for (int kk = 0; kk < 16; ++kk)
    gA(Cbf, arow, kk * 4 + koff, car[kk], cai[kk]);

  // --- x0: hash-based stand-in for the PRNG normal init in the reference
  v8f xr[2], xi[2];
  #pragma unroll
  for (int t = 0; t < 2; ++t) {
    int gn = (tj0 + t) * 16 + colL;
    #pragma unroll
    for (int v = 0; v < 8; ++v) {
      unsigned s = (unsigned)(b * 4096 + (rb + v) * 64 + gn);
      xr[t][v] = hash01(s * 2u + 1u) * 2.0f - 1.0f;
      xi[t][v] = hash01(s * 2u + 2u) * 2.0f - 1.0f;
    }
  }

  #pragma unroll 1
  for (int it = 0; it < ITERS; ++it) {
    // 1) publish x (transposed) to LDS
    #pragma unroll
    for (int t = 0; t < 2; ++t) {
      int gn = (tj0 + t) * 16 + colL;
      #pragma unroll
      for (int v = 0; v < 8; ++v) {
        XTre[gn][rb + v] = xr[t][v];
        XTim[gn][rb + v] = xi[t][v];
      }
    }
    __syncthreads();

    // 2) r = C @ x  (identity applied in step 3)
    v8f rr[2] = {z8, z8}, ri[2] = {z8, z8};
    #pragma unroll 4
    for (int kk = 0; kk < 16; ++kk) {
      const int k0 = kk * 4;
      v2f ar = car[kk], ai = cai[kk];
      v2f nai; nai.x = -ai.x; nai.y = -ai.y;
      #pragma unroll
      for (int t = 0; t < 2; ++t) {
        v2f br = ldsBT(XTre, k0, (tj0 + t) * 16, lane);
        v2f bi = ldsBT(XTim, k0, (tj0 + t) * 16, lane);
        rr[t] = wmma4(ar, br, rr[t]);
        rr[t] = wmma4(nai, bi, rr[t]);
        ri[t] = wmma4(ar, bi, ri[t]);
        ri[t] = wmma4(ai, br, ri[t]);
      }
    }
    __syncthreads();  // all waves done reading X

    // 3) P = (r - I)/|r - I| -> overwrite transposed planes
    #pragma unroll
    for (int t = 0; t < 2; ++t) {
      int gn = (tj0 + t) * 16 + colL;
      #pragma unroll
      for (int v = 0; v < 8; ++v) {
        int gm = rb + v;
        float a = rr[t][v] - ((gm == gn) ? 1.0f : 0.0f);
        float c = ri[t][v];
        float mag = sqrtf(a * a + c * c);
        float inv = mag > 0.0f ? 1.0f / mag : 0.0f;
        XTre[gn][gm] = a * inv;
        XTim[gn][gm] = c * inv;
      }
    }
    __syncthreads();

    // 4) G = C @ P ;  x -= scale * G
    v8f gr[2] = {z8, z8}, gi[2] = {z8, z8};
    #pragma unroll 4
    for (int kk = 0; kk < 16; ++kk) {
      const int k0 = kk * 4;
      v2f ar = car[kk], ai = cai[kk];
      v2f nai; nai.x = -ai.x; nai.y = -ai.y;
      #pragma unroll
      for (int t = 0; t < 2; ++t) {
        v2f br = ldsBT(XTre, k0, (tj0 + t) * 16, lane);
        v2f bi = ldsBT(XTim, k0, (tj0 + t) * 16, lane);
        gr[t] = wmma4(ar, br, gr[t]);
        gr[t] = wmma4(nai, bi, gr[t]);
        gi[t] = wmma4(ar, bi, gi[t]);
        gi[t] = wmma4(ai, br, gi[t]);
      }
    }
    #pragma unroll
    for (int t = 0; t < 2; ++t) {
      #pragma unroll
      for (int v = 0; v < 8; ++v) {
        xr[t][v] -= scale * gr[t][v];
        xi[t][v] -= scale * gi[t][v];
      }
    }
    __syncthreads();  // P reads done before next publish
  }

  float2* Ob = (float2*)Xout + (size_t)b * ND * ND;
  #pragma unroll
  for (int t = 0; t < 2; ++t) {
    int gn = (tj0 + t) * 16 + colL;
    #pragma unroll
    for (int v = 0; v < 8; ++v)
      Ob[(rb + v) * ND + gn] = make_float2(xr[t][v], xi[t][v]);
  }
}

// ===========================================================================
// Host launcher. Inputs: y[B,M,N] c64, no f32, C[B,N,N] c64, pilot c64.
// Output: h_hat[B,M,N] c64 then C_inv_approx[B,N,N] c64 (interleaved floats).
// ===========================================================================
extern "C" void kernel_launch(void* const* d_in, const int* in_sizes, int n_in,
                              void* d_out, int out_size, void* d_ws, size_t ws_size,
                              hipStream_t stream) {
  (void)in_sizes; (void)n_in; (void)out_size; (void)ws_size;
  const float* y     = (const float*)d_in[0];
  const float* no    = (const float*)d_in[1];
  const float* Cg    = (const float*)d_in[2];
  const float* pilot = (const float*)d_in[3];

  float* out  = (float*)d_out;
  float* hhat = out;                              // B*M*N complex
  float* cinv = out + (size_t)NB * MO * ND * 2;   // B*N*N complex

  float* ws   = (float*)d_ws;
  float* Ainv = ws;                               // B*N*N complex
  float* W    = ws + (size_t)NB * ND * ND * 2;    // B*N*N complex

  hipLaunchKernelGGL(k_herm_inverse, dim3(NB), dim3(64), 0, stream,
                     Cg, no, pilot, Ainv);
  hipLaunchKernelGGL(k_mmse_filter, dim3(NB), dim3(256), 0, stream,
                     Cg, Ainv, pilot, W);
  hipLaunchKernelGGL(k_apply_filter, dim3(NB, MO / 64), dim3(256), 0, stream,
                     y, W, hhat);
  const float scale = LR / (float)(NB * ND * ND);
  hipLaunchKernelGGL(k_gd_inverse, dim3(NB), dim3(256), 0, stream,
                     Cg, cinv, scale);
}